// PerLeadTimeLSTM_78383153152628
// MI455X (gfx1250) — compile-verified
//
#include <hip/hip_runtime.h>

#define LEADS   42
#define HID     10
#define ROWS    128                 // batch rows per block
#define THREADS 256                 // 8 waves (wave32)
#define NFROW   (LEADS * 2)         // 84 floats per batch row
#define TILE_F4 ((ROWS * NFROW) / 4) // 2688 float4s per tile

typedef __attribute__((ext_vector_type(2))) float v2f;
typedef __attribute__((ext_vector_type(8))) float v8f;

#define USE_ASYNC_LDS 1

// sigmoid(x) = 1/(1+e^-x) ; tanh(x) = 2/(1+e^-2x) - 1  (one exp + one rcp each)
__device__ __forceinline__ float act_sig_tanh(float x, bool isSig) {
  float arg = isSig ? -x : -2.0f * x;
  float e   = __expf(arg);                       // v_exp_f32
  float r   = __builtin_amdgcn_rcpf(1.0f + e);   // v_rcp_f32
  return isSig ? r : __builtin_fmaf(2.0f, r, -1.0f);
}

__global__ __launch_bounds__(THREADS, 2)
void lstm_lead_kernel(const float* __restrict__ x,
                      const float* __restrict__ W_ih,
                      const float* __restrict__ b_ih,
                      const float* __restrict__ b_hh,
                      const float* __restrict__ W_fc,
                      const float* __restrict__ b_fc,
                      float* __restrict__ y)
{
  __shared__ __align__(16) float lds_x[ROWS * NFROW]; // 43008 B: x in, y out (in place)
  __shared__ float wB[LEADS * 64];     // packed gate weights [t][p<32][k<2], f-gate dropped
  __shared__ float biasL[LEADS * 32];  // packed b_ih+b_hh for i,g,o (+2 zero pads)
  __shared__ float wfcL[LEADS * 20];   // W_fc[t][d][h] verbatim
  __shared__ float bfcL[LEADS * 2];
  __shared__ float hbuf[8 * 160];      // per-wave 16x10 h-transpose scratch

  const int tid = threadIdx.x;
  const int b0  = blockIdx.x * ROWS;

  // ---- pack weights into LDS (f-gate rows 10..19 of 4H skipped) ----
  for (int i = tid; i < LEADS * 32; i += THREADS) {
    int t = i >> 5, p = i & 31;
    float w0 = 0.f, w1 = 0.f, bb = 0.f;
    if (p < 30) {
      int orig = p + (p < 10 ? 0 : 10);          // i:0..9  g:20..29  o:30..39
      w0 = W_ih[t * 80 + orig * 2 + 0];
      w1 = W_ih[t * 80 + orig * 2 + 1];
      bb = b_ih[t * 40 + orig] + b_hh[t * 40 + orig];
    }
    wB[t * 64 + 2 * p + 0] = w0;
    wB[t * 64 + 2 * p + 1] = w1;
    biasL[i] = bb;
  }
  for (int i = tid; i < LEADS * 20; i += THREADS) wfcL[i] = W_fc[i];
  if (tid < LEADS * 2) bfcL[tid] = b_fc[tid];

  // ---- stage x tile into LDS, coalesced 16B/lane ----
#if USE_ASYNC_LDS
  {
    unsigned ldsbase = (unsigned)(size_t)(&lds_x[0]);
    unsigned long long gbase =
        (unsigned long long)((const char*)x + (size_t)b0 * NFROW * sizeof(float));
    for (int i = tid; i < TILE_F4; i += THREADS) {
      unsigned lo = ldsbase + (unsigned)i * 16u;
      unsigned vo = (unsigned)i * 16u;
      asm volatile("global_load_async_to_lds_b128 %0, %1, %2"
                   :: "v"(lo), "v"(vo), "s"(gbase) : "memory");
    }
    asm volatile("s_wait_asynccnt 0" ::: "memory");
  }
#else
  {
    const float4* src = (const float4*)(x + (size_t)b0 * NFROW);
    float4* dst = (float4*)lds_x;
    for (int i = tid; i < TILE_F4; i += THREADS) dst[i] = src[i];
  }
#endif
  __syncthreads();

  const int  wave   = tid >> 5;
  const int  lane   = tid & 31;
  const int  ln     = lane & 15;
  const bool loHalf = (lane < 16);

  for (int t = wave; t < LEADS; t += 8) {
    // B matrices (4x16 f32): VGPR0 = {K0 | K2}, VGPR1 = {K1 | K3}; K=2,3 are zero pad.
    v2f B0 = {0.f, 0.f}, B1 = {0.f, 0.f};
    if (loHalf) {
      B0.x = wB[t * 64 + 2 * ln + 0];
      B0.y = wB[t * 64 + 2 * ln + 1];
      B1.x = wB[t * 64 + 2 * (16 + ln) + 0];
      B1.y = wB[t * 64 + 2 * (16 + ln) + 1];
    }
    const float bias0 = biasL[t * 32 + ln];
    const float bias1 = biasL[t * 32 + 16 + ln];
    // packed column p: 0..9 = sigmoid(i_h), 10..19 = tanh(g_h), 20..29 = sigmoid(o_h)
    const bool sig0  = (ln < 10);                // D0 cols p = ln
    const bool sig1  = (ln >= 4) && (ln < 14);   // D1 cols p = 16+ln
    const int  srcTG = (lane & 16) + ((ln < 6) ? (10 + ln) : (ln - 6));
    const int  srcSO = (lane & 16) + 4 + ln;
    const int  mm = lane >> 1;                   // 16 rows x 2 outs = 32 lanes
    const int  dd = lane & 1;
    const float bfc_v = bfcL[t * 2 + dd];

    for (int rt = 0; rt < 8; ++rt) {
      // A matrix (16x4 f32): VGPR0 = {K0 | K2}, VGPR1 = {K1 | K3}; only K=0,1 real.
      v2f A = {0.f, 0.f};
      if (loHalf) {
        int row = rt * 16 + ln;
        A.x = lds_x[row * NFROW + 2 * t + 0];
        A.y = lds_x[row * NFROW + 2 * t + 1];
      }
      v8f C0, C1;
#pragma unroll
      for (int i = 0; i < 8; ++i) { C0[i] = bias0; C1[i] = bias1; }

      // gates(16x30) = x(16x2) @ Wpacked^T(2x30) + bias, full f32 WMMA
      v8f D0 = __builtin_amdgcn_wmma_f32_16x16x4_f32(false, A, false, B0, (short)0, C0, false, false);
      v8f D1 = __builtin_amdgcn_wmma_f32_16x16x4_f32(false, A, false, B1, (short)0, C1, false, false);

      float a0[8], a1[8];
#pragma unroll
      for (int v = 0; v < 8; ++v) {
        a0[v] = act_sig_tanh(D0[v], sig0);
        a1[v] = act_sig_tanh(D1[v], sig1);
      }
#pragma unroll
      for (int v = 0; v < 8; ++v) {
        float tgA = __shfl(a0[v], srcTG, 32);    // tanh(g_h) for h<6 (D0 lanes 10..15)
        float tgB = __shfl(a1[v], srcTG, 32);    // tanh(g_h) for h>=6 (D1 lanes 0..3)
        float tg  = (ln < 6) ? tgA : tgB;
        float so  = __shfl(a1[v], srcSO, 32);    // sigmoid(o_h) (D1 lanes 4..13)
        float c   = a0[v] * tg;                  // sigmoid(i_h) * tanh(g_h)
        float hv  = so * act_sig_tanh(c, false); // sigmoid(o) * tanh(c)
        if (ln < 10) {
          int m = v + (loHalf ? 0 : 8);          // D row = v (+8 for upper half)
          hbuf[wave * 160 + m * 10 + ln] = hv;
        }
      }
      __builtin_amdgcn_wave_barrier();           // wave-internal LDS W->R ordering

      // y[m][d] = b_fc[d] + sum_h h[m][h] * W_fc[d][h]   (lane = (m,d))
      float acc = bfc_v;
#pragma unroll
      for (int h = 0; h < HID; ++h)
        acc = __builtin_fmaf(hbuf[wave * 160 + mm * 10 + h],
                             wfcL[t * 20 + dd * 10 + h], acc);
      __builtin_amdgcn_wave_barrier();           // R->W before next tile reuses hbuf
      lds_x[(rt * 16 + mm) * NFROW + 2 * t + dd] = acc; // y in place of x
    }
  }
  __syncthreads();

  // ---- coalesced store of the whole y tile ----
  {
    float4* dst = (float4*)(y + (size_t)b0 * NFROW);
    const float4* src = (const float4*)lds_x;
    for (int i = tid; i < TILE_F4; i += THREADS) dst[i] = src[i];
  }
}

extern "C" void kernel_launch(void* const* d_in, const int* in_sizes, int n_in,
                              void* d_out, int out_size, void* d_ws, size_t ws_size,
                              hipStream_t stream) {
  const float* x    = (const float*)d_in[0];
  const float* W_ih = (const float*)d_in[1];
  const float* b_ih = (const float*)d_in[2];
  const float* b_hh = (const float*)d_in[3];
  const float* W_fc = (const float*)d_in[4];
  const float* b_fc = (const float*)d_in[5];
  float* y = (float*)d_out;

  int B = in_sizes[0] / NFROW;   // 131072
  int blocks = B / ROWS;         // 1024
  lstm_lead_kernel<<<blocks, THREADS, 0, stream>>>(x, W_ih, b_ih, b_hh, W_fc, b_fc, y);
}